// Baseline_62612033241506
// MI455X (gfx1250) — compile-verified
//
#include <hip/hip_runtime.h>
#include <cstdint>

// ---------------- problem constants (match reference) ----------------
#define B_     2048
#define N_     4098
#define K_     3073
#define M_     1025
#define E_     (N_ * 3)          // 12294
#define NITER_ 50
#define CAP_   64                // per-check-node edge capacity (Poisson(12), max ~30)
#define LLR_MAX_ 20.0f
#define PHI_EPS_ 8.5e-8f

// gfx1250 async-to-LDS path (probe-verified on this toolchain)
#if defined(__gfx1250__) && defined(__has_builtin)
#if __has_builtin(__builtin_amdgcn_global_load_async_to_lds_b32) && \
    __has_builtin(__builtin_amdgcn_s_wait_asynccnt)
#define USE_ASYNC_LDS 1
#endif
#endif

#if defined(USE_ASYNC_LDS)
typedef __attribute__((address_space(1))) int gint_as1;
typedef __attribute__((address_space(3))) int lint_as3;
#endif

// phi(x) = -log(tanh(x/2)), clipped like the reference; self-inverse.
__device__ __forceinline__ float phi_f(float x) {
  x = fminf(fmaxf(x, PHI_EPS_), LLR_MAX_);
  float em = __expf(-x);                          // e^-x
  float tb = __fdividef(1.0f - em, 1.0f + em);    // tanh(x/2)
  float t  = (x < 6.1e-5f) ? (0.5f * x) : tb;     // avoid 1-e^-x cancellation
  // log(t) <= 0; fabs == -log and converts the t==1 -> -0.0 case to +0.0
  return fabsf(__logf(t));
}

// pack sign(v2c) into the (strictly >= +0) phi magnitude
__device__ __forceinline__ float pack_msg(float v2c) {
  float ph = phi_f(fabsf(v2c));
  return (v2c < 0.0f) ? -ph : ph;
}

// reconstruct extrinsic c2v message from packed edge msg + packed CN aggregate
__device__ __forceinline__ float reconstruct_c2v(float tj, float A) {
  unsigned s  = __float_as_uint(tj) >> 31;   // edge sign
  unsigned pA = __float_as_uint(A) >> 31;    // CN parity
  float mag = phi_f(fabsf(A) - fabsf(tj));   // extrinsic magnitude
  return ((pA ^ s) & 1u) ? -mag : mag;
}

// ---------------- float4 helpers (B128 memory path) ----------------
__device__ __forceinline__ float4 ld4(const float* p) { return *(const float4*)p; }
__device__ __forceinline__ void   st4(float* p, const float4& v) { *(float4*)p = v; }
__device__ __forceinline__ float4 add4(const float4& a, const float4& b) {
  return make_float4(a.x + b.x, a.y + b.y, a.z + b.z, a.w + b.w);
}
__device__ __forceinline__ float4 sub4(const float4& a, const float4& b) {
  return make_float4(a.x - b.x, a.y - b.y, a.z - b.z, a.w - b.w);
}
__device__ __forceinline__ float4 rec4(const float4& tj, const float4& A) {
  return make_float4(reconstruct_c2v(tj.x, A.x), reconstruct_c2v(tj.y, A.y),
                     reconstruct_c2v(tj.z, A.z), reconstruct_c2v(tj.w, A.w));
}
__device__ __forceinline__ float4 pack4(const float4& v) {
  return make_float4(pack_msg(v.x), pack_msg(v.y), pack_msg(v.z), pack_msg(v.w));
}

// -------- llr [B,N] -> llrT [N,B], tiled transpose --------
__global__ __launch_bounds__(256) void k_transpose(const float* __restrict__ in,
                                                   float* __restrict__ out) {
  __shared__ float tile[32][33];
  int c0 = blockIdx.x * 32;   // column = variable node v
  int r0 = blockIdx.y * 32;   // row    = batch b
  for (int i = threadIdx.y; i < 32; i += 8) {
    int r = r0 + i, c = c0 + threadIdx.x;
    if (c < N_) tile[i][threadIdx.x] = in[r * N_ + c];
  }
  __syncthreads();
  for (int i = threadIdx.y; i < 32; i += 8) {
    int c = c0 + i, r = r0 + threadIdx.x;     // out[c*B + r]
    if (c < N_) out[c * B_ + r] = tile[threadIdx.x][i];
  }
}

// -------- deterministic per-check-node edge lists (ascending edge order) --------
__global__ __launch_bounds__(256) void k_build(const int* __restrict__ cn_idx,
                                               int* __restrict__ deg,
                                               int* __restrict__ lists) {
  const int cn  = blockIdx.x;
  const int tid = threadIdx.x;
  __shared__ int cnt_s[256];
  __shared__ int base_s[256];
  const int CH = (E_ + 255) / 256;            // 49
  int s = tid * CH;
  int e_end = s + CH; if (e_end > E_) e_end = E_;
  int c = 0;
  for (int e = s; e < e_end; ++e) c += (cn_idx[e] == cn);
  cnt_s[tid] = c;
  __syncthreads();
  if (tid == 0) {
    int acc = 0;
    for (int i = 0; i < 256; ++i) { base_s[i] = acc; acc += cnt_s[i]; }
    deg[cn] = acc;
  }
  __syncthreads();
  int pos = base_s[tid];
  for (int e = s; e < e_end; ++e)
    if (cn_idx[e] == cn) { if (pos < CAP_) lists[cn * CAP_ + pos] = e; ++pos; }
}

// -------- iteration 1: v2c = llr, pack into edge messages --------
__global__ __launch_bounds__(256) void k_init(const float* __restrict__ llrT,
                                              float* __restrict__ t) {
  const int v  = blockIdx.x;
  const int b4 = (blockIdx.y * 256 + threadIdx.x) * 4;
  const int base = 3 * v * B_ + b4;
  float4 p = pack4(ld4(llrT + v * B_ + b4));
  st4(t + base,           p);
  st4(t + base + B_,      p);
  st4(t + base + 2 * B_,  p);
}

// -------- CN aggregate: single gather pass, packed (sum, parity) per (cn,b) --------
__global__ __launch_bounds__(256) void k_cn_agg(const float* __restrict__ t,
                                                const int* __restrict__ deg,
                                                const int* __restrict__ lists,
                                                float* __restrict__ agg) {
  const int cn = blockIdx.x;
  const int b4 = (blockIdx.y * 256 + threadIdx.x) * 4;
  __shared__ int se[CAP_];
  int d = deg[cn]; if (d > CAP_) d = CAP_;
  if (threadIdx.x < (unsigned)d) {
#if defined(USE_ASYNC_LDS)
    __builtin_amdgcn_global_load_async_to_lds_b32(
        (gint_as1*)(lists + cn * CAP_ + threadIdx.x),
        (lint_as3*)&se[threadIdx.x],
        0, 0);
#else
    se[threadIdx.x] = lists[cn * CAP_ + threadIdx.x];
#endif
  }
#if defined(USE_ASYNC_LDS)
  __builtin_amdgcn_s_wait_asynccnt(0);
#endif
  __syncthreads();

  // get all rows in flight independent of the accumulation chain
  for (int k = 0; k < d; ++k)
    __builtin_prefetch(t + se[k] * B_ + b4, 0, 3);

  float sx = 0.f, sy = 0.f, sz = 0.f, sw = 0.f;
  unsigned px = 0u, py = 0u, pz = 0u, pw = 0u;
#pragma unroll 4
  for (int k = 0; k < d; ++k) {
    float4 tk = ld4(t + se[k] * B_ + b4);
    sx += fabsf(tk.x); px ^= (__float_as_uint(tk.x) >> 31);
    sy += fabsf(tk.y); py ^= (__float_as_uint(tk.y) >> 31);
    sz += fabsf(tk.z); pz ^= (__float_as_uint(tk.z) >> 31);
    sw += fabsf(tk.w); pw ^= (__float_as_uint(tk.w) >> 31);
  }
  float4 o = make_float4((px & 1u) ? -sx : sx,    // parity packed in sign bit
                         (py & 1u) ? -sy : sy,
                         (pz & 1u) ? -sz : sz,
                         (pw & 1u) ? -sw : sw);
  st4(agg + cn * B_ + b4, o);
}

// -------- VN update: reconstruct c2v from (t, agg), rewrite t in place --------
__global__ __launch_bounds__(256) void k_vn(const int* __restrict__ cn_idx,
                                            const float* __restrict__ agg,
                                            const float* __restrict__ llrT,
                                            float* __restrict__ t) {
  const int v  = blockIdx.x;
  const int b4 = (blockIdx.y * 256 + threadIdx.x) * 4;
  const int base = 3 * v * B_ + b4;
  float4 L = ld4(llrT + v * B_ + b4);
  // cn_idx[3v+j] is uniform per block -> scalar loads
  float4 c0 = rec4(ld4(t + base),          ld4(agg + cn_idx[3 * v]     * B_ + b4));
  float4 c1 = rec4(ld4(t + base + B_),     ld4(agg + cn_idx[3 * v + 1] * B_ + b4));
  float4 c2 = rec4(ld4(t + base + 2 * B_), ld4(agg + cn_idx[3 * v + 2] * B_ + b4));
  float4 s = add4(add4(add4(L, c0), c1), c2);
  st4(t + base,          pack4(sub4(s, c0)));
  st4(t + base + B_,     pack4(sub4(s, c1)));
  st4(t + base + 2 * B_, pack4(sub4(s, c2)));
}

// -------- posterior + hard decision + [N,B]->[B,K] transpose --------
__global__ __launch_bounds__(256) void k_out(const float* __restrict__ t,
                                             const float* __restrict__ agg,
                                             const int* __restrict__ cn_idx,
                                             const float* __restrict__ llrT,
                                             float* __restrict__ out) {
  __shared__ float tile[32][33];
  int v0 = blockIdx.x * 32;
  int b0 = blockIdx.y * 32;
  for (int i = threadIdx.y; i < 32; i += 8) {
    int v = v0 + i, b = b0 + threadIdx.x;
    if (v < K_) {
      float tot = llrT[v * B_ + b];
#pragma unroll
      for (int j = 0; j < 3; ++j) {
        int e = 3 * v + j;
        float tj = t[e * B_ + b];
        float A  = agg[cn_idx[e] * B_ + b];
        tot += reconstruct_c2v(tj, A);
      }
      tile[i][threadIdx.x] = tot;
    }
  }
  __syncthreads();
  for (int i = threadIdx.y; i < 32; i += 8) {
    int b = b0 + i, v = v0 + threadIdx.x;
    if (v < K_) out[b * K_ + v] = (tile[threadIdx.x][i] > 0.0f) ? 1.0f : 0.0f;
  }
}

extern "C" void kernel_launch(void* const* d_in, const int* in_sizes, int n_in,
                              void* d_out, int out_size, void* d_ws, size_t ws_size,
                              hipStream_t stream) {
  const float* llr    = (const float*)d_in[0];
  const int*   cn_idx = (const int*)d_in[1];
  // d_in[2] (vn_idx) is repeat(arange(N),3) by construction -> implicit.
  float* out = (float*)d_out;

  // workspace layout: t[E,B] | agg[M,B] | llrT[N,B] | deg[M] | lists[M*CAP]
  float* t     = (float*)d_ws;
  float* agg   = t    + (size_t)E_ * B_;
  float* llrT  = agg  + (size_t)M_ * B_;
  int*   deg   = (int*)(llrT + (size_t)N_ * B_);
  int*   lists = deg + M_;

  k_transpose<<<dim3((N_ + 31) / 32, B_ / 32), dim3(32, 8), 0, stream>>>(llr, llrT);
  k_build<<<M_, 256, 0, stream>>>(cn_idx, deg, lists);
  k_init<<<dim3(N_, B_ / 1024), 256, 0, stream>>>(llrT, t);

  for (int it = 0; it < NITER_; ++it) {
    k_cn_agg<<<dim3(M_, B_ / 1024), 256, 0, stream>>>(t, deg, lists, agg);
    if (it + 1 < NITER_)
      k_vn<<<dim3(N_, B_ / 1024), 256, 0, stream>>>(cn_idx, agg, llrT, t);
  }

  k_out<<<dim3((K_ + 31) / 32, B_ / 32), dim3(32, 8), 0, stream>>>(t, agg, cn_idx, llrT, out);
}